// AdaptivePseudoLabelRefinement_69801808494749
// MI455X (gfx1250) — compile-verified
//
#include <hip/hip_runtime.h>
#include <hip/hip_bf16.h>
#include <math.h>

// Problem constants (from reference shapes)
#define Qn 512
#define Dn 131072            // 512*16*16
#define Bn 2
#define An 16
#define Cn 19
#define HWn 65536            // 256*256
#define CONF 0.968f

#define TILES 32             // Qn / 16
#define SPLITS 256           // K splits for dist kernel (wave-level)
#define KCHUNK 512           // Dn / SPLITS
#define S2 64                // K splits for aug-dist kernel
#define KCHUNK2 2048         // Dn / S2

typedef __attribute__((ext_vector_type(2))) float v2f;
typedef __attribute__((ext_vector_type(8))) float v8f;

// ---------------------------------------------------------------------------
// K2: squared distances queue[512] vs tgt[2] via WMMA f32 16x16x4.
// A[m,k] = (q[m][k] - t[b][k])^2, B = all-ones  ->  D[m,n] = sum_k A[m,k]
// (row sums == dist^2 partials; B layout-agnostic since constant).
// grid (32 tiles, 32), block 256 (8 waves), wave split = by*8+wave.
// ---------------------------------------------------------------------------
__global__ void k2_dist_wmma(const float* __restrict__ queue,
                             const float* __restrict__ tgt,
                             float* __restrict__ dpart) {
    const int tile  = blockIdx.x;
    const int wave  = threadIdx.x >> 5;
    const int lane  = threadIdx.x & 31;
    const int split = blockIdx.y * 8 + wave;
    const int half  = lane >> 4;
    const int r     = lane & 15;

    const size_t row = (size_t)(tile * 16 + r);
    const int kbase  = split * KCHUNK + half * 2;
    const float* qp  = queue + row * (size_t)Dn + kbase;
    const float* t0p = tgt + kbase;
    const float* t1p = tgt + Dn + kbase;

    v8f c0 = {};
    v8f c1 = {};
    v2f ones; ones.x = 1.0f; ones.y = 1.0f;

    for (int kk = 0; kk < KCHUNK; kk += 4) {
        float2 q  = *(const float2*)(qp  + kk);
        float2 t0 = *(const float2*)(t0p + kk);
        float2 t1 = *(const float2*)(t1p + kk);
        __builtin_prefetch(qp + kk + 256, 0, 0);   // global_prefetch_b8
        v2f a0, a1;
        float d;
        d = q.x - t0.x; a0.x = d * d;
        d = q.y - t0.y; a0.y = d * d;
        d = q.x - t1.x; a1.x = d * d;
        d = q.y - t1.y; a1.y = d * d;
        c0 = __builtin_amdgcn_wmma_f32_16x16x4_f32(false, a0, false, ones,
                                                   (short)0, c0, false, false);
        c1 = __builtin_amdgcn_wmma_f32_16x16x4_f32(false, a1, false, ones,
                                                   (short)0, c1, false, false);
    }

    // C/D layout: VGPR j -> lanes 0-15: M=j,N=lane ; lanes 16-31: M=j+8.
    // All N columns identical (B==1), read column 0 from lanes 0 and 16.
    float* outp = dpart + (((size_t)tile * SPLITS + split) * 2) * 16;
    if (lane == 0) {
#pragma unroll
        for (int j = 0; j < 8; ++j) { outp[j]      = c0[j]; outp[16 + j]      = c1[j]; }
    }
    if (lane == 16) {
#pragma unroll
        for (int j = 0; j < 8; ++j) { outp[8 + j]  = c0[j]; outp[16 + 8 + j]  = c1[j]; }
    }
}

// ---------------------------------------------------------------------------
// K3: reduce split partials -> dist^2[512][2]; argmin per target (first-min).
// 1 block, 512 threads.
// ---------------------------------------------------------------------------
__global__ void k3_argmin(const float* __restrict__ dpart,
                          int* __restrict__ imin,
                          float* __restrict__ cd2) {
    __shared__ float sd[512];
    __shared__ int   si[512];
    const int m = threadIdx.x;
    const int tile = m >> 4, r = m & 15;

    const float* base = dpart + (size_t)tile * SPLITS * 32;  // [split][2][16]
    float s0 = 0.0f, s1 = 0.0f;
    for (int sp = 0; sp < SPLITS; ++sp) {
        s0 += base[sp * 32 + r];
        s1 += base[sp * 32 + 16 + r];
    }

    for (int b = 0; b < 2; ++b) {
        __syncthreads();
        sd[m] = (b == 0) ? s0 : s1;
        si[m] = m;
        __syncthreads();
        for (int off = 256; off > 0; off >>= 1) {
            if (m < off) {
                float dv = sd[m + off]; int di = si[m + off];
                if (dv < sd[m] || (dv == sd[m] && di < si[m])) { sd[m] = dv; si[m] = di; }
            }
            __syncthreads();
        }
        if (m == 0) { imin[b] = si[0]; cd2[b] = sd[0]; }
    }
}

// ---------------------------------------------------------------------------
// K4: d2^2[b][a] partials: || auged[b][a] - queue[imin[b]] ||^2
// grid (32, S2), block 256.
// ---------------------------------------------------------------------------
__global__ void k4_augdist(const float* __restrict__ queue,
                           const float* __restrict__ auged,
                           const int* __restrict__ imin,
                           float* __restrict__ part2) {
    const int ba = blockIdx.x;            // b*16 + a
    const int b = ba >> 4;
    const int chunk = blockIdx.y;
    const int row = imin[b];
    const float* c = queue + (size_t)row * Dn + chunk * KCHUNK2;
    const float* x = auged + (size_t)ba * Dn + chunk * KCHUNK2;

    float s = 0.0f;
    for (int j = threadIdx.x; j < KCHUNK2; j += 256) {
        float d = x[j] - c[j];
        s += d * d;
    }
    __shared__ float red[256];
    red[threadIdx.x] = s;
    __syncthreads();
    for (int off = 128; off > 0; off >>= 1) {
        if (threadIdx.x < off) red[threadIdx.x] += red[threadIdx.x + off];
        __syncthreads();
    }
    if (threadIdx.x == 0) part2[ba * S2 + chunk] = red[0];
}

// ---------------------------------------------------------------------------
// K5: finish d2, masked k-smallest selection (ties -> lowest index).
// 1 block, 32 threads.
// ---------------------------------------------------------------------------
__global__ void k5_select(const float* __restrict__ part2,
                          const float* __restrict__ cd2,
                          const int* __restrict__ kptr,
                          int* __restrict__ selIdx,
                          int* __restrict__ selN) {
    __shared__ float d2s[32];
    const int i = threadIdx.x;
    float s = 0.0f;
    for (int j = 0; j < S2; ++j) s += part2[i * S2 + j];
    d2s[i] = s;
    __syncthreads();

    if (i < 2) {
        const int b = i;
        const float cd = cd2[b];
        int kk = *kptr;
        if (kk < 1) kk = 1;
        if (kk > An) kk = An;
        bool taken[An];
#pragma unroll
        for (int a = 0; a < An; ++a) taken[a] = false;
        int n = 0;
        for (int t = 0; t < kk; ++t) {
            int best = -1; float bv = 0.0f;
            for (int a = 0; a < An; ++a) {
                float v = d2s[b * An + a];
                if (taken[a] || v > cd) continue;   // keep v <= cd (squared == norm order)
                if (best < 0 || v < bv) { best = a; bv = v; }
            }
            if (best < 0) break;
            taken[best] = true;
            selIdx[b * An + n] = best;
            ++n;
        }
        selN[b] = n;
        for (int t = n; t < An; ++t) selIdx[b * An + t] = 0;
    }
}

// ---------------------------------------------------------------------------
// K6: weighted softmax average over selected augs, argmax, confidence gate.
// grid (HW/256, B), block 256; one thread per pixel.
// ---------------------------------------------------------------------------
__global__ void k6_final(const float* __restrict__ auged_logits,
                         const float* __restrict__ tgt_logits,
                         const int* __restrict__ plbl,
                         const int* __restrict__ selIdx,
                         const int* __restrict__ selN,
                         int* __restrict__ out_refined,
                         float* __restrict__ out_avg) {
    const int px = blockIdx.x * 256 + threadIdx.x;
    const int b  = blockIdx.y;
    const int n  = selN[b];

    float avg[Cn];
#pragma unroll
    for (int c = 0; c < Cn; ++c) avg[c] = 0.0f;

    for (int s = 0; s < n; ++s) {
        const int aidx = selIdx[b * An + s];
        const float* base =
            auged_logits + (((size_t)b * An + aidx) * Cn) * HWn + px;
        float l[Cn];
        float m = -INFINITY;
#pragma unroll
        for (int c = 0; c < Cn; ++c) { l[c] = base[(size_t)c * HWn]; m = fmaxf(m, l[c]); }
        float sum = 0.0f;
#pragma unroll
        for (int c = 0; c < Cn; ++c) { l[c] = expf(l[c] - m); sum += l[c]; }
        const float inv = 1.0f / sum;
#pragma unroll
        for (int c = 0; c < Cn; ++c) avg[c] += l[c] * inv;
    }

    const float wnorm = 1.0f / fmaxf((float)n, 1.0f);
#pragma unroll
    for (int c = 0; c < Cn; ++c) avg[c] *= wnorm;

    // argmax over classes (first max wins, like jnp.argmax)
    int knn = 0; float bv = avg[0];
#pragma unroll
    for (int c = 1; c < Cn; ++c) { if (avg[c] > bv) { bv = avg[c]; knn = c; } }

    const int pl = plbl[b * HWn + px];
    const int label = (n > 0) ? knn : pl;

    // max softmax of target logits: exp(0)/sum(exp(x-max)) = 1/sum
    const float* tb = tgt_logits + ((size_t)b * Cn) * HWn + px;
    float m2 = -INFINITY;
    float tl[Cn];
#pragma unroll
    for (int c = 0; c < Cn; ++c) { tl[c] = tb[(size_t)c * HWn]; m2 = fmaxf(m2, tl[c]); }
    float s2 = 0.0f;
#pragma unroll
    for (int c = 0; c < Cn; ++c) s2 += expf(tl[c] - m2);
    const bool low_conf = (1.0f / s2) < CONF;

    out_refined[b * HWn + px] = low_conf ? label : pl;
#pragma unroll
    for (int c = 0; c < Cn; ++c)
        out_avg[((size_t)b * Cn + c) * HWn + px] = avg[c];
}

// ---------------------------------------------------------------------------
extern "C" void kernel_launch(void* const* d_in, const int* in_sizes, int n_in,
                              void* d_out, int out_size, void* d_ws, size_t ws_size,
                              hipStream_t stream) {
    const float* queue        = (const float*)d_in[0];  // [512, 131072]
    const float* tgt          = (const float*)d_in[1];  // [2, 131072]
    const float* tgt_logits   = (const float*)d_in[2];  // [2, 19, 65536]
    const float* auged        = (const float*)d_in[3];  // [2, 16, 131072]
    const float* auged_logits = (const float*)d_in[4];  // [2, 16, 19, 65536]
    const int*   plbl         = (const int*)d_in[5];    // [2, 65536]
    const int*   kptr         = (const int*)d_in[6];    // scalar k

    // workspace layout (floats/ints, ~1.07 MB total)
    float* dpart  = (float*)d_ws;                       // [32][256][2][16]
    float* part2  = dpart + (size_t)TILES * SPLITS * 2 * 16;  // [32][64]
    int*   imin   = (int*)(part2 + 32 * S2);            // [2]
    float* cd2    = (float*)(imin + 2);                 // [2]
    int*   selIdx = (int*)(cd2 + 2);                    // [2][16]
    int*   selN   = selIdx + 2 * An;                    // [2]

    k2_dist_wmma<<<dim3(TILES, SPLITS / 8), 256, 0, stream>>>(queue, tgt, dpart);
    k3_argmin<<<1, 512, 0, stream>>>(dpart, imin, cd2);
    k4_augdist<<<dim3(Bn * An, S2), 256, 0, stream>>>(queue, auged, imin, part2);
    k5_select<<<1, 32, 0, stream>>>(part2, cd2, kptr, selIdx, selN);

    int*   out_refined = (int*)d_out;                   // [2][65536] int32 bits
    float* out_avg     = (float*)d_out + Bn * HWn;      // [2][19][65536]
    k6_final<<<dim3(HWn / 256, Bn), 256, 0, stream>>>(auged_logits, tgt_logits,
                                                      plbl, selIdx, selN,
                                                      out_refined, out_avg);
}